// HexSmoothSparse_38448547234487
// MI455X (gfx1250) — compile-verified
//
#include <hip/hip_runtime.h>

// HexSmoothSparse: out[i,:] = mean over 7 gathered neighbour rows of x (fp32, F=128).
// Memory-bound gather (AI ~ 0.85 flop/byte). One wave32 per output row; one lane
// owns 4 contiguous floats (global_load_b128 -> 512B fully-coalesced per row per
// neighbour). Row index is wave-uniform (readfirstlane) so neighbour indices ride
// the scalar path (s_load / KMcnt) and each gather uses GVS addressing:
// SGPR64 row base + shared 32-bit VGPR lane offset. x (84MB) stays resident in the
// 192MB L2 across the 7x gather reuse; output is stored non-temporally so the
// 84MB write stream doesn't evict it. HBM floor ~172MB ~= 7.4us @ 23.3TB/s.

typedef float v4f __attribute__((ext_vector_type(4)));

#define N_FEAT 128
#define N_NEIGH 7
#define WAVES_PER_BLOCK 8   // 256 threads = 8 wave32

__global__ __launch_bounds__(256)
void HexSmoothSparse_gather_mean_kernel(const float* __restrict__ x,
                                        const int*   __restrict__ neighbours,
                                        float*       __restrict__ out,
                                        int n_points)
{
    const int lane = (int)(threadIdx.x & 31u);
    const int row  = (int)blockIdx.x * WAVES_PER_BLOCK + (int)(threadIdx.x >> 5);
    if (row >= n_points) return;   // wave-uniform guard: EXEC stays all-ones

    // Force uniformity so the index loads scalarize (s_load, KMcnt) and all
    // index->address math lands on the SALU.
    const int rowU = __builtin_amdgcn_readfirstlane(row);

    const int* __restrict__ nr = neighbours + (size_t)rowU * N_NEIGH;
    const unsigned j0 = (unsigned)nr[0];
    const unsigned j1 = (unsigned)nr[1];
    const unsigned j2 = (unsigned)nr[2];
    const unsigned j3 = (unsigned)nr[3];
    const unsigned j4 = (unsigned)nr[4];
    const unsigned j5 = (unsigned)nr[5];
    const unsigned j6 = (unsigned)nr[6];

    // Divergent component: 32-bit byte offset lane*16 (shared by all 7 loads).
    const size_t voff = (size_t)((unsigned)lane * 16u);
    const char* __restrict__ xb = (const char*)x;

    // 7 independent global_load_b128 (SGPR base + VGPR offset), latencies overlap.
    const v4f a0 = *(const v4f*)(xb + ((size_t)j0 << 9) + voff);
    const v4f a1 = *(const v4f*)(xb + ((size_t)j1 << 9) + voff);
    const v4f a2 = *(const v4f*)(xb + ((size_t)j2 << 9) + voff);
    const v4f a3 = *(const v4f*)(xb + ((size_t)j3 << 9) + voff);
    const v4f a4 = *(const v4f*)(xb + ((size_t)j4 << 9) + voff);
    const v4f a5 = *(const v4f*)(xb + ((size_t)j5 << 9) + voff);
    const v4f a6 = *(const v4f*)(xb + ((size_t)j6 << 9) + voff);

    v4f s = ((a0 + a1) + (a2 + a3)) + ((a4 + a5) + a6);
    s = s * (1.0f / 7.0f);   // matches jnp.mean (sum * 1/K)

    // Non-temporal store: keep the gather-hot x resident in L2.
    v4f* dst = (v4f*)((char*)out + ((size_t)(unsigned)rowU << 9) + voff);
    __builtin_nontemporal_store(s, dst);
}

extern "C" void kernel_launch(void* const* d_in, const int* in_sizes, int n_in,
                              void* d_out, int out_size, void* d_ws, size_t ws_size,
                              hipStream_t stream)
{
    const float* x          = (const float*)d_in[0];
    const int*   neighbours = (const int*)d_in[1];
    float*       out        = (float*)d_out;

    const int n_points = in_sizes[1] / N_NEIGH;   // 163842
    const int blocks   = (n_points + WAVES_PER_BLOCK - 1) / WAVES_PER_BLOCK;

    hipLaunchKernelGGL(HexSmoothSparse_gather_mean_kernel,
                       dim3(blocks), dim3(256), 0, stream,
                       x, neighbours, out, n_points);
}